// PostProcessing2_3590592659709
// MI455X (gfx1250) — compile-verified
//
#include <hip/hip_runtime.h>

// ---------------------------------------------------------------------------
// Panoptic post-processing for gfx1250 (MI455X).
// WMMA f32 16x16x4 accelerates the 3x streaming-argmin center assignment:
//   score(p,c) = -2*(px*cx+py*cy) + (cx^2+cy^2)   (== dist^2 - |p|^2)
// A-matrix rows = centers (-2cx, -2cy, cx^2+cy^2, 0), B cols = pixels (px,py,1,0).
// A-operand is fetched branch-free: lanes 0-15 read the (x,y) half of the
// float4 table entry, lanes 16-31 read the (z,w)=(c^2,0) half -> one
// global_load_b64 per tile, all 13 tiles batched before the WMMA chain.
// ---------------------------------------------------------------------------

namespace {
constexpr int Hc = 512, Wc = 1024, HWc = Hc * Wc;
constexpr int PADc = 7;
constexpr int Hp = Hc + 2 * PADc, Wp = Wc + 2 * PADc, HWp = Hp * Wp;
constexpr int NCc = 34;
constexpr int Bc = 2;
constexpr int TOPK = 200, K1 = TOPK + 1;
constexpr int KP = 208;                 // centers padded to 13 WMMA tiles
constexpr int NTILE = KP / 16;
constexpr int CAP = 4096;               // candidate list capacity per batch
constexpr float BIGV = 1e18f;           // sentinel for invalid centers
}

typedef float v2f __attribute__((ext_vector_type(2)));
typedef float v8f __attribute__((ext_vector_type(8)));

// ---- per-pixel: class argmax, things mask, center votes ----
__global__ void k_perpixel(const float* __restrict__ sp, const float* __restrict__ cr,
                           float* __restrict__ out_seg, unsigned char* __restrict__ segc,
                           unsigned char* __restrict__ things, unsigned int* __restrict__ vote)
{
    int gid = blockIdx.x * blockDim.x + threadIdx.x;
    if (gid >= Bc * HWc) return;
    int b = gid / HWc, pix = gid - b * HWc;
    const float* p = sp + (size_t)b * NCc * HWc + pix;
    float mv = p[0]; int mc = 0;
    for (int c = 1; c < NCc; ++c) {
        float v = p[(size_t)c * HWc];
        if (v > mv) { mv = v; mc = c; }
    }
    segc[gid] = (unsigned char)mc;
    out_seg[gid] = (float)mc;
    int th = (mc >= 24) && (mc <= 33);              // seg_map in (23.99, 33]
    things[gid] = (unsigned char)th;
    int y = pix / Wc, x = pix - y * Wc;
    float ccx = (float)(x + 1) - cr[(size_t)b * 2 * HWc + pix];
    float ccy = (float)(y + 1) - cr[(size_t)b * 2 * HWc + HWc + pix];
    int vx = (int)rintf(ccx), vy = (int)rintf(ccy); // round-half-even == jnp.round
    if (th && vx >= 0 && vy >= 0 && vx < Wc && vy < Hc)
        atomicAdd(&vote[b * HWc + vy * Wc + vx], 1u);
}

// ---- 7x7 max-pool NMS on zero-padded vote map; emit candidates (>50) ----
__global__ void k_nms(const unsigned int* __restrict__ vote,
                      unsigned int* __restrict__ cand_val, int* __restrict__ cand_idx,
                      int* __restrict__ cand_cnt)
{
    int gid = blockIdx.x * blockDim.x + threadIdx.x;
    if (gid >= Bc * HWp) return;
    int b = gid / HWp, r = gid - b * HWp;
    int yp = r / Wp, xp = r - yp * Wp;
    int y0 = yp - PADc, x0 = xp - PADc;
    unsigned int a = 0;
    if ((unsigned)y0 < (unsigned)Hc && (unsigned)x0 < (unsigned)Wc)
        a = vote[b * HWc + y0 * Wc + x0];
    unsigned int m = 0;
    for (int dy = -3; dy <= 3; ++dy) {
        int yy = y0 + dy;
        if ((unsigned)yy >= (unsigned)Hc) continue;
        for (int dx = -3; dx <= 3; ++dx) {
            int xx = x0 + dx;
            if ((unsigned)xx >= (unsigned)Wc) continue;
            unsigned int v = vote[b * HWc + yy * Wc + xx];
            m = (v > m) ? v : m;
        }
    }
    if (m == a && a > 50u) {                        // pooled==aggr && aggr>50
        int pos = atomicAdd(&cand_cnt[b], 1);
        if (pos < CAP) { cand_val[b * CAP + pos] = a; cand_idx[b * CAP + pos] = r; }
    }
}

// ---- deterministic top-K selection (value desc, index asc), one block/batch ----
__global__ void k_topk(unsigned int* __restrict__ cand_val, const int* __restrict__ cand_idx,
                       const int* __restrict__ cand_cnt, float* __restrict__ centers,
                       unsigned char* __restrict__ cvalid, int* __restrict__ nvalid,
                       int* __restrict__ remap)
{
    __shared__ unsigned long long skey[256];
    __shared__ int spos[256];
    __shared__ int s_nval;
    int b = blockIdx.x, t = threadIdx.x;
    int n = cand_cnt[b]; if (n > CAP) n = CAP;
    unsigned int* cv = cand_val + b * CAP;
    const int* ci = cand_idx + b * CAP;
    if (t == 0) s_nval = 0;
    for (int k = 0; k < TOPK; ++k) {
        unsigned long long bk = 0ull; int bp = -1;
        for (int i = t; i < n; i += 256) {
            unsigned int v = cv[i];
            if (v) {
                unsigned long long key =
                    ((unsigned long long)v << 32) |
                    (unsigned long long)(0xFFFFFFFFu - (unsigned int)ci[i]);
                if (key > bk) { bk = key; bp = i; }
            }
        }
        skey[t] = bk; spos[t] = bp;
        __syncthreads();
        for (int s = 128; s > 0; s >>= 1) {
            if (t < s && skey[t + s] > skey[t]) { skey[t] = skey[t + s]; spos[t] = spos[t + s]; }
            __syncthreads();
        }
        if (t == 0) {
            int base = (b * TOPK + k) * 2;
            if (skey[0] != 0ull) {
                int pos = spos[0];
                int idx = ci[pos];
                centers[base + 0] = (float)(idx / Wp);   // cy (padded coords)
                centers[base + 1] = (float)(idx % Wp);   // cx
                cvalid[b * TOPK + k] = 1;
                cv[pos] = 0u;                             // mark used
                s_nval++;
            } else {
                centers[base + 0] = 0.f; centers[base + 1] = 0.f;
                cvalid[b * TOPK + k] = 0;
            }
        }
        __syncthreads();
    }
    if (t == 0) nvalid[b] = s_nval;
    for (int j = t; j <= TOPK; j += 256) remap[b * K1 + j] = j;   // identity for pass 0
}

// ---- build WMMA A-matrix table from centers + per-pass validity ----
__global__ void k_ctab(const float* __restrict__ centers, const unsigned char* __restrict__ valid,
                       float4* __restrict__ ctab)
{
    int gid = blockIdx.x * blockDim.x + threadIdx.x;
    if (gid >= Bc * KP) return;
    int b = gid / KP, k = gid - b * KP;
    float4 o;
    if (k < TOPK && valid[b * TOPK + k]) {
        float cy = centers[(b * TOPK + k) * 2 + 0];
        float cx = centers[(b * TOPK + k) * 2 + 1];
        o = make_float4(-2.f * cx, -2.f * cy, cx * cx + cy * cy, 0.f);
    } else {
        o = make_float4(0.f, 0.f, BIGV, 0.f);
    }
    ctab[b * KP + k] = o;
}

// ---- WMMA argmin assignment: one wave handles 16 pixels x 208 centers ----
__global__ void k_assign(const float* __restrict__ cr, const float4* __restrict__ ctab,
                         const int* __restrict__ nvalid, const int* __restrict__ remap,
                         const int* __restrict__ prevlab, const unsigned char* __restrict__ things,
                         int* __restrict__ outlab)
{
    int tid = blockIdx.x * blockDim.x + threadIdx.x;
    int wid = tid >> 5;
    int lane = tid & 31;
    constexpr int WPB = HWc / 16;                  // waves per batch (32768, exact)
    if (wid >= Bc * WPB) return;
    int b = wid / WPB;
    int lp = lane & 15;
    int hi = lane & 16;                            // 0 for lanes 0-15, 16 for 16-31
    int pix = (wid - b * WPB) * 16 + lp;
    int gpix = b * HWc + pix;

    if (nvalid[b] == 0) {                          // wave-uniform fallback branch
        if (lane < 16) outlab[gpix] = prevlab ? prevlab[gpix] : 0;
        return;
    }

    // B matrix (4x16, K x pixels): lanes 0-15 hold K=0,1 = (px,py); 16-31 hold K=2,3 = (1,0)
    v2f bv;
    if (lane < 16) {
        int y = pix / Wc, x = pix - (pix / Wc) * Wc;
        bv[0] = (float)(x + 1) - cr[(size_t)b * 2 * HWc + pix];
        bv[1] = (float)(y + 1) - cr[(size_t)b * 2 * HWc + HWc + pix];
    } else {
        bv[0] = 1.0f; bv[1] = 0.0f;
    }

    // Branch-free batched A-operand loads: one global_load_b64 per tile.
    // lanes 0-15 -> (x,y) half of float4; lanes 16-31 -> (z,w)=(c^2,0) half.
    const char* abase = (const char*)(ctab + (size_t)b * KP + lp) + (hi ? 8 : 0);
    v2f avs[NTILE];
    #pragma unroll
    for (int t = 0; t < NTILE; ++t)
        avs[t] = *(const v2f*)(abase + (size_t)t * 256);   // 16 centers * 16B per tile

    float bestv = 3.0e38f;
    int besti = 0;
    int kb0 = hi ? 8 : 0;
    #pragma unroll
    for (int t = 0; t < NTILE; ++t) {
        v8f acc = {};
        acc = __builtin_amdgcn_wmma_f32_16x16x4_f32(false, avs[t], false, bv,
                                                    (short)0, acc, false, false);
        // D: lane<16 -> pixel lp, centers t*16+0..7 ; lane>=16 -> pixel lp, centers t*16+8..15
        int kb = t * 16 + kb0;
        #pragma unroll
        for (int r = 0; r < 8; ++r) {
            float vv = acc[r];
            if (vv < bestv) { bestv = vv; besti = kb + r; }   // strict < : earliest k wins
        }
    }
    float ov = __shfl_xor(bestv, 16, 32);
    int oi = __shfl_xor(besti, 16, 32);
    if (ov < bestv || (ov == bestv && oi < besti)) { bestv = ov; besti = oi; }
    if (lane < 16) {
        int lab = 0;
        if (things[gpix] && besti < TOPK) lab = remap[b * K1 + besti + 1];
        outlab[gpix] = lab;
    }
}

// ---- pass-1 stats: count + bbox per label (int atomics, deterministic) ----
__global__ void k_stats1(const int* __restrict__ lab, unsigned int* __restrict__ cnt,
                         int* __restrict__ ymin, int* __restrict__ ymax,
                         int* __restrict__ xmin, int* __restrict__ xmax)
{
    int gid = blockIdx.x * blockDim.x + threadIdx.x;
    if (gid >= Bc * HWc) return;
    int l = lab[gid];
    if (l <= 0) return;
    int b = gid / HWc, pix = gid - b * HWc;
    int y = pix / Wc, x = pix - y * Wc;
    int s = b * K1 + l;
    atomicAdd(&cnt[s], 1u);
    atomicMin(&ymin[s], y); atomicMax(&ymax[s], y);
    atomicMin(&xmin[s], x); atomicMax(&xmax[s], x);
}

__global__ void k_g1(const unsigned int* __restrict__ cnt, const int* __restrict__ ymin,
                     const int* __restrict__ ymax, const int* __restrict__ xmin,
                     const int* __restrict__ xmax, unsigned char* __restrict__ g,
                     int* __restrict__ remap, int* __restrict__ nvalid)
{
    int b = blockIdx.x * blockDim.x + threadIdx.x;
    if (b >= Bc) return;
    int run = 0;
    remap[b * K1] = 0;
    for (int k = 1; k <= TOPK; ++k) {
        int s = b * K1 + k;
        unsigned int c = cnt[s];
        int gg = 0;
        if (c > 0u) {
            float area = (float)(ymax[s] - ymin[s] + 1) * (float)(xmax[s] - xmin[s] + 1);
            gg = ((float)c / area) > 0.3f;
        }
        g[b * TOPK + k - 1] = (unsigned char)gg;
        run += gg;
        remap[s] = run;                             // == nl1[k-1]
    }
    nvalid[b] = run;
}

// ---- pass-2 stats: count + sums of center_regressions ----
__global__ void k_stats2(const int* __restrict__ lab, const float* __restrict__ cr,
                         unsigned int* __restrict__ cnt2, float* __restrict__ sx,
                         float* __restrict__ sy)
{
    int gid = blockIdx.x * blockDim.x + threadIdx.x;
    if (gid >= Bc * HWc) return;
    int l = lab[gid];
    if (l <= 0) return;
    int b = gid / HWc, pix = gid - b * HWc;
    int s = b * K1 + l;
    atomicAdd(&cnt2[s], 1u);
    atomicAdd(&sx[s], cr[(size_t)b * 2 * HWc + pix]);
    atomicAdd(&sy[s], cr[(size_t)b * 2 * HWc + HWc + pix]);
}

__global__ void k_g2(const unsigned int* __restrict__ cnt2, const float* __restrict__ sx,
                     const float* __restrict__ sy, unsigned char* __restrict__ g,
                     int* __restrict__ remap, int* __restrict__ nvalid)
{
    int b = blockIdx.x * blockDim.x + threadIdx.x;
    if (b >= Bc) return;
    int run = 0;
    remap[b * K1] = 0;
    for (int k = 1; k <= TOPK; ++k) {
        int s = b * K1 + k;
        float c = fmaxf((float)cnt2[s], 1.f);
        float mx = sx[s] / c, my = sy[s] / c;
        int gg = (cnt2[s] > 0u) && (fabsf(mx) < 10.f) && (fabsf(my) < 10.f);
        g[b * TOPK + k - 1] = (unsigned char)gg;
        run += gg;
        remap[s] = run;
    }
    nvalid[b] = run;
}

// ---- final stats: class histogram, counts; emit final labels as float ----
__global__ void k_stats3(const int* __restrict__ lab, const unsigned char* __restrict__ segc,
                         unsigned int* __restrict__ hist, unsigned int* __restrict__ cntu,
                         float* __restrict__ out_final)
{
    int gid = blockIdx.x * blockDim.x + threadIdx.x;
    if (gid >= Bc * HWc) return;
    int b = gid / HWc;
    int l = lab[gid];
    atomicAdd(&hist[(b * K1 + l) * NCc + segc[gid]], 1u);
    atomicAdd(&cntu[b * K1 + l], 1u);
    out_final[gid] = (float)l;
}

__global__ void k_g3(const unsigned int* __restrict__ hist, const unsigned int* __restrict__ cntu,
                     int* __restrict__ instc, float* __restrict__ out_inst,
                     float* __restrict__ out_cnt)
{
    int s = blockIdx.x * blockDim.x + threadIdx.x;
    if (s >= Bc * K1) return;
    const unsigned int* h = hist + (size_t)s * NCc;
    unsigned int mv = h[0]; int mc = 0;
    for (int c = 1; c < NCc; ++c) if (h[c] > mv) { mv = h[c]; mc = c; }
    instc[s] = mc;
    out_inst[s] = (float)mc;
    out_cnt[s] = (float)cntu[s];
}

// ---- per-pixel softmax prob at instance class, summed per instance ----
__global__ void k_probsum(const float* __restrict__ sp, const int* __restrict__ lab,
                          const int* __restrict__ instc, float* __restrict__ Sg)
{
    int gid = blockIdx.x * blockDim.x + threadIdx.x;
    if (gid >= Bc * HWc) return;
    int b = gid / HWc, pix = gid - b * HWc;
    int l = lab[gid];
    int c = instc[b * K1 + l];
    const float* p = sp + (size_t)b * NCc * HWc + pix;
    float v[NCc];
    float mx = -3.4e38f;
    for (int i = 0; i < NCc; ++i) { v[i] = p[(size_t)i * HWc]; mx = fmaxf(mx, v[i]); }
    float se = 0.f;
    for (int i = 0; i < NCc; ++i) se += __expf(v[i] - mx);
    float pr = __expf(v[c] - mx) / se;
    atomicAdd(&Sg[b * K1 + l], pr);
}

__global__ void k_g4(const float* __restrict__ Sg, const unsigned int* __restrict__ cntu,
                     float* __restrict__ out_prob)
{
    int s = blockIdx.x * blockDim.x + threadIdx.x;
    if (s >= Bc * K1) return;
    out_prob[s] = Sg[s] / fmaxf((float)cntu[s], 1.f);
}

// ---------------------------------------------------------------------------
extern "C" void kernel_launch(void* const* d_in, const int* in_sizes, int n_in,
                              void* d_out, int out_size, void* d_ws, size_t ws_size,
                              hipStream_t stream)
{
    (void)in_sizes; (void)n_in; (void)out_size; (void)ws_size;
    const float* sp = (const float*)d_in[0];      // [B,NC,H,W]
    const float* cr = (const float*)d_in[2];      // [B,2,H,W]
    float* out = (float*)d_out;
    float* out_final = out;
    float* out_seg   = out + Bc * HWc;
    float* out_inst  = out + 2 * Bc * HWc;
    float* out_prob  = out_inst + Bc * K1;
    float* out_cnt   = out_prob + Bc * K1;

    char* w = (char*)d_ws;
    auto take = [&](size_t bytes) -> void* {
        void* p = (void*)w;
        w += (bytes + 255) & ~(size_t)255;
        return p;
    };
    unsigned int* vote     = (unsigned int*)take((size_t)Bc * HWc * 4);
    unsigned char* things  = (unsigned char*)take((size_t)Bc * HWc);
    unsigned char* segc    = (unsigned char*)take((size_t)Bc * HWc);
    int* lab0              = (int*)take((size_t)Bc * HWc * 4);
    int* lab1              = (int*)take((size_t)Bc * HWc * 4);
    int* lab2              = (int*)take((size_t)Bc * HWc * 4);
    unsigned int* cand_val = (unsigned int*)take((size_t)Bc * CAP * 4);
    int* cand_idx          = (int*)take((size_t)Bc * CAP * 4);
    int* cand_cnt          = (int*)take((size_t)Bc * 4);
    float* centers         = (float*)take((size_t)Bc * TOPK * 2 * 4);
    unsigned char* cvalid  = (unsigned char*)take((size_t)Bc * TOPK);
    unsigned char* gbuf    = (unsigned char*)take((size_t)Bc * TOPK);
    float4* ctab           = (float4*)take((size_t)Bc * KP * 16);
    int* nvalid            = (int*)take((size_t)Bc * 4);
    int* remap             = (int*)take((size_t)Bc * K1 * 4);
    unsigned int* cnt      = (unsigned int*)take((size_t)Bc * K1 * 4);
    int* ymin              = (int*)take((size_t)Bc * K1 * 4);
    int* ymax              = (int*)take((size_t)Bc * K1 * 4);
    int* xmin              = (int*)take((size_t)Bc * K1 * 4);
    int* xmax              = (int*)take((size_t)Bc * K1 * 4);
    unsigned int* cnt2     = (unsigned int*)take((size_t)Bc * K1 * 4);
    float* sx              = (float*)take((size_t)Bc * K1 * 4);
    float* sy              = (float*)take((size_t)Bc * K1 * 4);
    unsigned int* hist     = (unsigned int*)take((size_t)Bc * K1 * NCc * 4);
    unsigned int* cntu     = (unsigned int*)take((size_t)Bc * K1 * 4);
    float* Sg              = (float*)take((size_t)Bc * K1 * 4);
    int* instc             = (int*)take((size_t)Bc * K1 * 4);

    // zero/init scratch (graph-capture-legal stream memsets)
    hipMemsetAsync(vote, 0, (size_t)Bc * HWc * 4, stream);
    hipMemsetAsync(cand_cnt, 0, (size_t)Bc * 4, stream);
    hipMemsetAsync(cnt, 0, (size_t)Bc * K1 * 4, stream);
    hipMemsetAsync(ymin, 0x7F, (size_t)Bc * K1 * 4, stream);
    hipMemsetAsync(ymax, 0, (size_t)Bc * K1 * 4, stream);
    hipMemsetAsync(xmin, 0x7F, (size_t)Bc * K1 * 4, stream);
    hipMemsetAsync(xmax, 0, (size_t)Bc * K1 * 4, stream);
    hipMemsetAsync(cnt2, 0, (size_t)Bc * K1 * 4, stream);
    hipMemsetAsync(sx, 0, (size_t)Bc * K1 * 4, stream);
    hipMemsetAsync(sy, 0, (size_t)Bc * K1 * 4, stream);
    hipMemsetAsync(hist, 0, (size_t)Bc * K1 * NCc * 4, stream);
    hipMemsetAsync(cntu, 0, (size_t)Bc * K1 * 4, stream);
    hipMemsetAsync(Sg, 0, (size_t)Bc * K1 * 4, stream);

    const int BLK = 256;
    int nPix = Bc * HWc;
    int gPix = (nPix + BLK - 1) / BLK;
    int nPad = Bc * HWp;
    int gPad = (nPad + BLK - 1) / BLK;
    int nAsg = (Bc * (HWc / 16)) * 32;             // 32 threads per 16-pixel wave
    int gAsg = (nAsg + BLK - 1) / BLK;
    int gCt  = (Bc * KP + BLK - 1) / BLK;
    int gK1  = (Bc * K1 + BLK - 1) / BLK;

    // 1) per-pixel class argmax + votes
    k_perpixel<<<gPix, BLK, 0, stream>>>(sp, cr, out_seg, segc, things, vote);
    // 2) NMS + candidate extraction
    k_nms<<<gPad, BLK, 0, stream>>>(vote, cand_val, cand_idx, cand_cnt);
    // 3) top-K (deterministic), identity remap, nvalid0
    k_topk<<<Bc, 256, 0, stream>>>(cand_val, cand_idx, cand_cnt, centers, cvalid, nvalid, remap);
    // 4) pass 0 assignment (WMMA)
    k_ctab<<<gCt, BLK, 0, stream>>>(centers, cvalid, ctab);
    k_assign<<<gAsg, BLK, 0, stream>>>(cr, ctab, nvalid, remap, nullptr, things, lab0);
    // 5) bbox-ratio pruning
    k_stats1<<<gPix, BLK, 0, stream>>>(lab0, cnt, ymin, ymax, xmin, xmax);
    k_g1<<<1, 32, 0, stream>>>(cnt, ymin, ymax, xmin, xmax, gbuf, remap, nvalid);
    k_ctab<<<gCt, BLK, 0, stream>>>(centers, gbuf, ctab);
    k_assign<<<gAsg, BLK, 0, stream>>>(cr, ctab, nvalid, remap, lab0, things, lab1);
    // 6) high-mean-regression pruning
    k_stats2<<<gPix, BLK, 0, stream>>>(lab1, cr, cnt2, sx, sy);
    k_g2<<<1, 32, 0, stream>>>(cnt2, sx, sy, gbuf, remap, nvalid);
    k_ctab<<<gCt, BLK, 0, stream>>>(centers, gbuf, ctab);
    k_assign<<<gAsg, BLK, 0, stream>>>(cr, ctab, nvalid, remap, lab1, things, lab2);
    // 7) instance class / size / mean prob
    k_stats3<<<gPix, BLK, 0, stream>>>(lab2, segc, hist, cntu, out_final);
    k_g3<<<gK1, BLK, 0, stream>>>(hist, cntu, instc, out_inst, out_cnt);
    k_probsum<<<gPix, BLK, 0, stream>>>(sp, lab2, instc, Sg);
    k_g4<<<gK1, BLK, 0, stream>>>(Sg, cntu, out_prob);
}